// BilateralSolverLocal3D_88270167867422
// MI455X (gfx1250) — compile-verified
//
#include <hip/hip_runtime.h>

// ---------------------------------------------------------------------------
// BilateralSolverLocal3D loss for MI455X (gfx1250, wave32).
//
// Memory-bound: streams 282 MB of w_ij once (NT-hinted b128 loads);
// output/target (128 KB each) stay L2-resident for the clamped gathers.
// WMMA f32 16x16x4 (B = ones) performs the 64-accumulator wave reduction.
// ---------------------------------------------------------------------------

typedef float v2f __attribute__((ext_vector_type(2)));
typedef float v4f __attribute__((ext_vector_type(4)));
typedef float v8f __attribute__((ext_vector_type(8)));

namespace {
constexpr int KOFF = 2204;   // 21*21*5 - 1 offsets
constexpr int NPIX = 32000;  // 5*80*80
constexpr int DT   = 5;
constexpr int DH   = 80;
constexpr int DW   = 80;
constexpr int CENTER_LIN = (10 * 21 + 10) * 5 + 2;  // 1102, skipped offset
}

// Reduce the two per-lane accumulators of every lane in the wave with one
// v_wmma_f32_16x16x4_f32: A holds the 64 accumulator values, B = ones, so
// D[m,n] = sum_k A[m,k] (columns identical). Sum the 8 D VGPRs per lane,
// fold lanes 0-15 with 16-31 via xor-16 shuffle -> every lane has the wave
// total. Then 8 wave totals are combined through LDS.
// Requires EXEC == all ones (guaranteed: no divergent exits in callers).
__device__ __forceinline__ float block_reduce_wmma(float acc0, float acc1,
                                                   float* lds8) {
  v2f a;  a.x = acc0; a.y = acc1;
  v2f bo; bo.x = 1.0f; bo.y = 1.0f;
  v8f c = {};
  v8f d = __builtin_amdgcn_wmma_f32_16x16x4_f32(
      /*neg_a=*/false, a, /*neg_b=*/false, bo,
      /*c_mod=*/(short)0, c, /*reuse_a=*/false, /*reuse_b=*/false);
  float s = d[0] + d[1] + d[2] + d[3] + d[4] + d[5] + d[6] + d[7];
  s += __shfl_xor(s, 16, 32);  // fold rows 0-7 with rows 8-15

  const int lane = threadIdx.x & 31;
  const int wid  = threadIdx.x >> 5;
  if (lane == 0) lds8[wid] = s;
  __syncthreads();
  float tot = 0.0f;
  if (threadIdx.x == 0) {
#pragma unroll
    for (int i = 0; i < 8; ++i) tot += lds8[i];
  }
  return tot;  // valid in thread 0
}

// One block per offset plane (b < KOFF), block KOFF does the fidelity term.
// 256 threads = 8 wave32s. Writes raw (unscaled) partial sums to part[].
__global__ __launch_bounds__(256)
void bilateral3d_partials(const float* __restrict__ w_ij,
                          const float* __restrict__ target,
                          const float* __restrict__ outp,
                          float* __restrict__ part) {
  const int b   = blockIdx.x;
  const int tid = threadIdx.x;
  __shared__ float lds8[8];

  float acc0 = 0.0f, acc1 = 0.0f;
  const v4f* __restrict__ out4 = reinterpret_cast<const v4f*>(outp);

  if (b < KOFF) {
    // Recover (dt,dh,dw) from flat index, skipping the center offset.
    const int orig = b + (b >= CENTER_LIN ? 1 : 0);
    const int k  = orig % 5;
    const int ij = orig / 5;
    const int j  = ij % 21;
    const int i  = ij / 21;
    const int dt = k - 2, dh = i - 10, dw = j - 10;

    const float* __restrict__ wq  = w_ij + (size_t)b * NPIX;
    const v4f*   __restrict__ wq4 = reinterpret_cast<const v4f*>(wq);

    // 31 float4 iterations (31744 px) + one scalar tail pass (256 px).
    // 80 % 4 == 0, so each float4 stays inside one (t,h) row.
#pragma unroll 2
    for (int it = 0; it < 31; ++it) {
      const int p  = it * 256 + tid;  // float4 index
      const int pb = p << 2;          // first pixel of the quad
      const v4f wv = __builtin_nontemporal_load(wq4 + p);
      const v4f cv = out4[p];
      const int t = pb / (DH * DW);
      const int r = pb - t * (DH * DW);
      const int h = r / DW;
      const int x = r - h * DW;
      const int tn = min(max(t + dt, 0), DT - 1);
      const int hn = min(max(h + dh, 0), DH - 1);
      const int nb = (tn * DH + hn) * DW;
#pragma unroll
      for (int u = 0; u < 4; ++u) {
        const int xn = min(max(x + u + dw, 0), DW - 1);
        const float dv = cv[u] - outp[nb + xn];
        const float v  = wv[u] * dv * dv;
        if (u & 1) acc1 += v; else acc0 += v;
      }
    }
    {  // tail: pixels 31744..31999 (uniform across the block)
      const int p = 31 * 1024 + tid;
      const int t = p / (DH * DW);
      const int r = p - t * (DH * DW);
      const int h = r / DW;
      const int x = r - h * DW;
      const int tn = min(max(t + dt, 0), DT - 1);
      const int hn = min(max(h + dh, 0), DH - 1);
      const int xn = min(max(x + dw, 0), DW - 1);
      const float dv = outp[p] - outp[(tn * DH + hn) * DW + xn];
      acc0 += __builtin_nontemporal_load(wq + p) * dv * dv;
    }
  } else {
    // Fidelity partial: sum (out - target)^2.
    const v4f* __restrict__ tgt4 = reinterpret_cast<const v4f*>(target);
#pragma unroll 2
    for (int it = 0; it < 31; ++it) {
      const int p  = it * 256 + tid;
      const v4f cv = out4[p];
      const v4f tv = tgt4[p];
#pragma unroll
      for (int u = 0; u < 4; ++u) {
        const float e = cv[u] - tv[u];
        if (u & 1) acc1 += e * e; else acc0 += e * e;
      }
    }
    {
      const int p = 31 * 1024 + tid;
      const float e = outp[p] - target[p];
      acc0 += e * e;
    }
  }

  const float tot = block_reduce_wmma(acc0, acc1, lds8);
  if (threadIdx.x == 0) part[b] = tot;
}

// Deterministic final reduction of the 2205 block partials.
//   loss = (LAM/K) * sum(smooth partials) + (1/NPIX) * fidelity partial
__global__ __launch_bounds__(256)
void bilateral3d_finish(const float* __restrict__ part,
                        float* __restrict__ outv) {
  __shared__ float sm[256];
  const int tid = threadIdx.x;
  float acc = 0.0f;
  for (int p = tid; p < KOFF; p += 256) acc += part[p];
  acc *= (128.0f / (float)KOFF);          // n_pix*LAM*mean -> LAM/K * sum
  if (tid == 0) acc += part[KOFF] * (1.0f / (float)NPIX);
  sm[tid] = acc;
  __syncthreads();
#pragma unroll
  for (int s = 128; s > 0; s >>= 1) {
    if (tid < s) sm[tid] += sm[tid + s];
    __syncthreads();
  }
  if (tid == 0) outv[0] = sm[0];
}

extern "C" void kernel_launch(void* const* d_in, const int* in_sizes, int n_in,
                              void* d_out, int out_size, void* d_ws, size_t ws_size,
                              hipStream_t stream) {
  const float* w_ij   = (const float*)d_in[0];  // (2204, 5, 80, 80) f32
  const float* target = (const float*)d_in[1];  // (5, 80, 80) f32
  const float* output = (const float*)d_in[2];  // (5, 80, 80) f32
  float* out  = (float*)d_out;                  // scalar f32
  float* part = (float*)d_ws;                   // 2205 partials

  bilateral3d_partials<<<KOFF + 1, 256, 0, stream>>>(w_ij, target, output, part);
  bilateral3d_finish<<<1, 256, 0, stream>>>(part, out);
}